// CDLoss_11381663334570
// MI455X (gfx1250) — compile-verified
//
#include <hip/hip_runtime.h>
#include <hip/hip_bf16.h>

// Chamfer distance on MI455X (gfx1250) via f32 WMMA 16x16x4.
//
// d2(p,g) = p.p - 2 p.g + g.g is produced by ONE wmma with no per-tile VALU
// fixup:
//   A row (loop-invariant)   = (px, py, pz, 1)
//   B col (pre-packed)       = (-2gx, -2gy, -2gz, g.g)
//   C accumulator (invariant)= p.p broadcast per row (built with 8 shfl)
//   => D[m][n] = d2(p_m, g_n) exactly (f32, matches reference precision)
// Clamp-at-0 commutes with min, so it is applied once after the scan.
//
// Hot loop per 16x16 tile (256 pairs): 1 coalesced b64 load + 1 v_wmma +
// ~8 v_min (v_min3 pairs under 2x unroll). Both chamfer directions run in
// one launch (d2 symmetric). Deterministic: partial sums to ws, LDS-tree
// finalize, no float atomics.

typedef __attribute__((ext_vector_type(2))) float v2f;
typedef __attribute__((ext_vector_type(8))) float v8f;

#define NPOINTS 8192
#define NTILES  (NPOINTS / 16)      // 512
#define NBATCH  4
#define NDIR    2
#define WAVES_PER_BLOCK 8           // 256 threads = 8 wave32

// ws layout: [0, 16KB)  float wsum[NDIR][NBATCH][NTILES]
//            [16KB, +1MB) v2f packB[NDIR][NBATCH][NTILES][32]
#define WSUM_FLOATS   (NDIR * NBATCH * NTILES)            // 4096
#define PACKB_BYTES   ((size_t)NDIR * NBATCH * NTILES * 32 * 8)   // 1 MB
#define WS_REQUIRED   (WSUM_FLOATS * 4 + PACKB_BYTES)

// Build the per-lane WMMA B operand for every 16-point tile of both clouds.
// packB segment 0 <- gt (columns for dir 0), segment 1 <- pred.
__global__ __launch_bounds__(256)
void chamfer_pack_kernel(const float* __restrict__ pred,
                         const float* __restrict__ gt,
                         v2f* __restrict__ packB)
{
    const int idx  = blockIdx.x * 256 + threadIdx.x;   // 0 .. 131071
    const int lane = idx & 31;
    const int tg   = idx >> 5;                         // (dir,b,mt) flat
    const int dir  = tg >> 11;
    const int b    = (tg >> 9) & (NBATCH - 1);
    const int mt   = tg & (NTILES - 1);

    const float* src = (dir == 0) ? gt : pred;
    const float* q = src + ((size_t)b * NPOINTS + mt * 16 + (lane & 15)) * 3;
    const float gx = q[0], gy = q[1], gz = q[2];
    const float gsq = gx * gx + gy * gy + gz * gz;

    v2f o;                                   // B 4x16 layout (ISA 7.12.2):
    if (lane < 16) { o.x = -2.0f * gx; o.y = -2.0f * gy; }  // K=0,1
    else           { o.x = -2.0f * gz; o.y = gsq;        }  // K=2,3
    packB[idx] = o;
}

__global__ __launch_bounds__(256)
void chamfer_main_kernel(const float* __restrict__ pred,
                         const float* __restrict__ gt,
                         const v2f* __restrict__ packB,   // may be null
                         float* __restrict__ wsum)
{
    const int lane  = threadIdx.x & 31;
    const int gwave = blockIdx.x * WAVES_PER_BLOCK + (threadIdx.x >> 5);
    const int dir   = gwave >> 11;               // 0: pred rows, 1: gt rows
    const int b     = (gwave >> 9) & (NBATCH - 1);
    const int tile  = gwave & (NTILES - 1);
    const int sub   = lane & 15;
    const bool hi   = lane >= 16;

    const float* X = ((dir == 0) ? pred : gt) + (size_t)b * NPOINTS * 3;
    const float* Y = ((dir == 0) ? gt : pred) + (size_t)b * NPOINTS * 3;

    // ---- loop-invariant A operand: row M = sub, A = (px,py,pz,1) ----
    const float* prow = X + (size_t)(tile * 16 + sub) * 3;
    const float px = prow[0], py = prow[1], pz = prow[2];
    const float psq = px * px + py * py + pz * pz;
    v2f a;
    a.x = hi ? pz   : px;     // lanes 0-15: K=0 ; lanes 16-31: K=2
    a.y = hi ? 1.0f : py;     // lanes 0-15: K=1 ; lanes 16-31: K=3

    // ---- loop-invariant C: C[m][n] = p.p of row m (D-layout broadcast) ----
    v8f cbias;
    const int srcBase = hi ? 8 : 0;
#pragma unroll
    for (int i = 0; i < 8; ++i)
        cbias[i] = __shfl(psq, srcBase + i, 32);

    v8f rmin;
#pragma unroll
    for (int i = 0; i < 8; ++i) rmin[i] = 3.4e38f;

    if (packB) {
        const v2f* pb = packB + (((size_t)dir * NBATCH + b) * NTILES) * 32 + lane;
        for (int mt = 0; mt < NTILES; ++mt, pb += 32) {
            const v2f bb = *pb;                       // coalesced b64
            v8f c = __builtin_amdgcn_wmma_f32_16x16x4_f32(
                false, a, false, bb, (short)0, cbias, false, false);
#pragma unroll
            for (int i = 0; i < 8; ++i) rmin[i] = fminf(rmin[i], c[i]);
        }
    } else {
        for (int mt = 0; mt < NTILES; ++mt) {
            const float* grow = Y + (size_t)(mt * 16 + sub) * 3;
            const float gx = grow[0], gy = grow[1], gz = grow[2];
            const float gsq = gx * gx + gy * gy + gz * gz;
            v2f bb;
            bb.x = hi ? (-2.0f * gz) : (-2.0f * gx);
            bb.y = hi ? gsq          : (-2.0f * gy);
            v8f c = __builtin_amdgcn_wmma_f32_16x16x4_f32(
                false, a, false, bb, (short)0, cbias, false, false);
#pragma unroll
            for (int i = 0; i < 8; ++i) rmin[i] = fminf(rmin[i], c[i]);
        }
    }

    // Min across the 16 columns held by each lane half, clamp once, sum.
    float s = 0.0f;
#pragma unroll
    for (int i = 0; i < 8; ++i) {
        float v = rmin[i];
        v = fminf(v, __shfl_xor(v, 1, 32));
        v = fminf(v, __shfl_xor(v, 2, 32));
        v = fminf(v, __shfl_xor(v, 4, 32));
        v = fminf(v, __shfl_xor(v, 8, 32));
        s += fmaxf(v, 0.0f);                 // clamp commutes with min
    }
    const float otherHalf = __shfl_xor(s, 16, 32);   // rows 8-15 half
    if (lane == 0)
        wsum[((size_t)dir * NBATCH + b) * NTILES + tile] = s + otherHalf;
}

__global__ __launch_bounds__(256)
void chamfer_finalize_kernel(const float* __restrict__ wsum,
                             float* __restrict__ out)
{
    __shared__ float sm[256];
    const int b = blockIdx.x;
    const int t = threadIdx.x;
    float s = 0.0f;
    for (int i = t; i < NTILES; i += 256)
        s += wsum[(size_t)b * NTILES + i]                      // dir 0
           + wsum[((size_t)NBATCH + b) * NTILES + i];          // dir 1
    sm[t] = s;
    __syncthreads();
    for (int w = 128; w >= 1; w >>= 1) {
        if (t < w) sm[t] += sm[t + w];
        __syncthreads();
    }
    if (t == 0) out[b] = sm[0] / (float)NPOINTS;   // N == M == 8192
}

extern "C" void kernel_launch(void* const* d_in, const int* in_sizes, int n_in,
                              void* d_out, int out_size, void* d_ws, size_t ws_size,
                              hipStream_t stream) {
    (void)in_sizes; (void)n_in; (void)out_size;
    const float* pred = (const float*)d_in[0];   // [4, 8192, 3] f32
    const float* gt   = (const float*)d_in[1];   // [4, 8192, 3] f32
    float* out  = (float*)d_out;                 // [4] f32
    float* wsum = (float*)d_ws;                  // 4096 floats

    const bool usePacked = (ws_size >= WS_REQUIRED);
    v2f* packB = usePacked ? (v2f*)((char*)d_ws + WSUM_FLOATS * 4) : nullptr;

    const dim3 block(256);
    if (usePacked) {
        // 2 dirs * 4 batches * 512 tiles * 32 lanes = 131072 threads
        chamfer_pack_kernel<<<dim3(512), block, 0, stream>>>(pred, gt, packB);
    }
    // 2 dirs * 4 batches * 512 tiles = 4096 waves = 512 blocks of 8 waves
    chamfer_main_kernel<<<dim3(512), block, 0, stream>>>(pred, gt, packB, wsum);
    chamfer_finalize_kernel<<<dim3(NBATCH), block, 0, stream>>>(wsum, out);
}